// SPGG_Qlearning_85504208929335
// MI455X (gfx1250) — compile-verified
//
#include <hip/hip_runtime.h>

typedef unsigned int u32x4 __attribute__((ext_vector_type(4)));
typedef int          i32x4 __attribute__((ext_vector_type(4)));
typedef int          i32x8 __attribute__((ext_vector_type(8)));
typedef float        f32x4 __attribute__((ext_vector_type(4)));
typedef float        f32x2 __attribute__((ext_vector_type(2)));

#define LSIDE 4096
#define LMASK 4095
#define BX 64
#define BY 4
#define NTHREADS (BX * BY)

#if __has_builtin(__builtin_amdgcn_tensor_load_to_lds) && __has_builtin(__builtin_amdgcn_s_wait_tensorcnt)
#define USE_TDM 1
#else
#define USE_TDM 0
#endif

#if __has_builtin(__builtin_amdgcn_global_load_async_to_lds_b32) && __has_builtin(__builtin_amdgcn_s_wait_asynccnt)
#define USE_ASYNC 1
#else
#define USE_ASYNC 0
#endif

__global__ __launch_bounds__(NTHREADS) void spgg_qstep(
    const int*   __restrict__ type_t,     // (L,L) int32, values 0/1
    const float* __restrict__ q_tab,      // (N,2,2) f32
    const float* __restrict__ rand_t,     // (N,2) f32
    const int*   __restrict__ rand_type,  // (L,L) int32
    const float* __restrict__ eps_rand,   // (L,L) f32
    float* __restrict__ out_type,         // N
    float* __restrict__ out_q,            // 4N
    float* __restrict__ out_profit)       // N
{
    __shared__ int   sT[BY + 4][BX + 4];                 // type tile, halo 2
    __shared__ float sC[BY + 2][BX + 4];                 // coop_num, halo 1 (padded width)
    __shared__ __attribute__((aligned(16))) float sQ[BY * BX * 4];  // q tile via TDM

    const int tx  = threadIdx.x;          // 0..63
    const int ty  = threadIdx.y;          // 0..3
    const int tid = ty * BX + tx;         // 0..255
    const int bx  = blockIdx.x * BX;
    const int by  = blockIdx.y * BY;

#if USE_TDM
    if (tid < 32) {  // wave 0 issues one TDM DMA for the whole block's q tile
        unsigned long long ga =
            (unsigned long long)(const void*)(q_tab + ((unsigned long long)by * LSIDE + bx) * 4ull);
        unsigned lds_base = (unsigned)(unsigned long long)(const void*)&sQ[0]; // low 32 = LDS offset

        u32x4 g0;
        g0[0] = 1u;                                   // count=1 (valid), user mode, no gather
        g0[1] = lds_base;                             // lds_addr
        g0[2] = (unsigned)ga;                         // global_addr[31:0]
        g0[3] = (unsigned)((ga >> 32) & 0x01FFFFFFull) | (2u << 30); // addr[56:32] | type=2

        i32x8 g1;
        g1[0] = (int)(2u << 16);                      // data_size=4B; no multicast/pad/iterate
        g1[1] = (int)((16384u & 0xFFFFu) << 16);      // tensor_dim0 lo16 (row = 16384 floats)
        g1[2] = (int)((16384u >> 16) | ((4096u & 0xFFFFu) << 16)); // dim0 hi | tensor_dim1 lo
        g1[3] = (int)((4096u >> 16) | (256u << 16));  // dim1 hi | tile_dim0 = 256 elements
        g1[4] = 4;                                    // tile_dim1 = 4 rows, tile_dim2 = 0
        g1[5] = 16384;                                // tensor_dim0_stride lo32 (elements)
        g1[6] = 0;                                    // stride hi16 | dim1_stride lo16
        g1[7] = 0;

        i32x4 gz4 = {0, 0, 0, 0};                     // 2D tensor: groups 2/3 unused
        i32x8 gz8 = {0, 0, 0, 0, 0, 0, 0, 0};        // extra group (6-arg toolchain form)
        __builtin_amdgcn_tensor_load_to_lds(g0, g1, gz4, gz4, gz8, 0);
    }
#endif

    // ---- stage type tile with halo 2, torus wrap ----
    // gfx1250 async global->LDS path (no VGPR round-trip, overlaps with TDM DMA)
    for (int i = tid; i < (BY + 4) * (BX + 4); i += NTHREADS) {
        int lx = i % (BX + 4);
        int ly = i / (BX + 4);
        int gx = (bx + lx - 2) & LMASK;
        int gy = (by + ly - 2) & LMASK;
        const int* gp = &type_t[gy * LSIDE + gx];
#if USE_ASYNC
        __builtin_amdgcn_global_load_async_to_lds_b32((int*)gp, &sT[ly][lx], 0, 0);
#else
        sT[ly][lx] = *gp;
#endif
    }
#if USE_ASYNC
    __builtin_amdgcn_s_wait_asynccnt(0);
#endif
    __syncthreads();

    // ---- coop_num = neigh5(c) over (BX+2)x(BY+2) inner-halo region ----
    for (int i = tid; i < (BY + 2) * (BX + 2); i += NTHREADS) {
        int cx = i % (BX + 2);
        int cy = i / (BX + 2);
        int s = sT[cy + 1][cx + 1] + sT[cy][cx + 1] + sT[cy + 2][cx + 1]
              + sT[cy + 1][cx]     + sT[cy + 1][cx + 2];
        sC[cy][cx] = (float)s;
    }
#if USE_TDM
    __builtin_amdgcn_s_wait_tensorcnt(0);   // no-op for waves with TENSORcnt==0
#endif
    __syncthreads();                        // publish sC and TDM-written sQ to all waves

    // ---- per-cell: profit + epsilon-greedy + Q update ----
    const int gy = by + ty, gx = bx + tx;
    const long long cell = (long long)gy * LSIDE + gx;

    const int A = sT[ty + 2][tx + 2];       // current state (0/1)
    float S = sC[ty + 1][tx + 1] + sC[ty][tx + 1] + sC[ty + 2][tx + 1]
            + sC[ty + 1][tx]     + sC[ty + 1][tx + 2];
    // profit = neigh5(c_profit)*c + neigh5(d_profit)*d == (R/5)*S - 5*c
    const float profit = 0.96f * S - (A == 1 ? 5.0f : 0.0f);

#if USE_TDM
    const f32x4 qv = *(const f32x4*)&sQ[(ty * BX + tx) * 4];
#else
    const f32x4 qv = __builtin_nontemporal_load((const f32x4*)(q_tab + cell * 4));
#endif
    const f32x2 rv = __builtin_nontemporal_load((const f32x2*)(rand_t + cell * 2));
    const float er = __builtin_nontemporal_load(eps_rand + cell);
    const int   rt = __builtin_nontemporal_load(rand_type + cell);

    // epsilon-greedy with random tie-break (argmax takes first max -> strict > for index 1)
    const float Qp0 = A ? qv.z : qv.x;
    const float Qp1 = A ? qv.w : qv.y;
    const float mx  = fmaxf(Qp0, Qp1);
    const float m0  = (Qp0 == mx) ? 1.0f : 0.0f;
    const float m1  = (Qp1 == mx) ? 1.0f : 0.0f;
    const float s0  = m0 - (1.0f - m0) * 1.0e9f + rv.x;
    const float s1  = m1 - (1.0f - m1) * 1.0e9f + rv.y;
    const int greedy = (s1 > s0) ? 1 : 0;
    const int B = (er >= 0.02f) ? greedy : rt;

    // Q[s,A,B] <- (1-eta)*Q[s,A,B] + eta*(profit + gamma*max_a' Q[s,B,a'])
    const float QB0  = B ? qv.z : qv.x;
    const float QB1  = B ? qv.w : qv.y;
    const float maxB = fmaxf(QB0, QB1);
    const float qAB  = A ? (B ? qv.w : qv.z) : (B ? qv.y : qv.x);
    const float ETA = 0.8f, GAMMA = 0.8f;
    const float upd = (1.0f - ETA) * qAB + ETA * (profit + GAMMA * maxB);

    f32x4 qo = qv;
    qo[A * 2 + B] = upd;

    // streaming (non-temporal) outputs: never re-read, keep L2 for the type matrix
    __builtin_nontemporal_store((float)B, out_type + cell);
    __builtin_nontemporal_store(qo, (f32x4*)(out_q + cell * 4));
    __builtin_nontemporal_store(profit, out_profit + cell);
}

extern "C" void kernel_launch(void* const* d_in, const int* in_sizes, int n_in,
                              void* d_out, int out_size, void* d_ws, size_t ws_size,
                              hipStream_t stream) {
    (void)in_sizes; (void)n_in; (void)out_size; (void)d_ws; (void)ws_size;
    const int*   type_t    = (const int*)  d_in[0];
    const float* q_tab     = (const float*)d_in[1];
    const float* rand_t    = (const float*)d_in[2];
    const int*   rand_type = (const int*)  d_in[3];
    const float* eps_rand  = (const float*)d_in[4];

    const long long N = (long long)LSIDE * LSIDE;
    float* out        = (float*)d_out;
    float* out_type   = out;           // type_t1_matrix : N
    float* out_q      = out + N;       // q_new          : 4N
    float* out_profit = out + 5 * N;   // profit         : N

    dim3 block(BX, BY);
    dim3 grid(LSIDE / BX, LSIDE / BY);
    spgg_qstep<<<grid, block, 0, stream>>>(type_t, q_tab, rand_t, rand_type, eps_rand,
                                           out_type, out_q, out_profit);
}